// pointpillars_78993038508100
// MI455X (gfx1250) — compile-verified
//
#include <hip/hip_runtime.h>

#ifndef INT_MAX
#define INT_MAX 2147483647
#endif

// Grid derived from COORS_RANGE / VOXEL_SIZE in the reference
#define GX 256
#define GY 128
#define GZ 1
#define NUM_CELLS (GX * GY * GZ)   // 32768
#define MAX_POINTS 100
#define MAX_VOXELS 12000
#define CHUNK_SIZE 8192
#define MAX_CHUNKS 256                              // supports N <= 2,097,152
#define WORDS_MAX ((MAX_CHUNKS * CHUNK_SIZE) / 64)  // 32768 u64 words

// Flat output layout (return-order concat): voxels | num_points | coors | voxel_num
#define VOX_ELEMS  (MAX_VOXELS * MAX_POINTS * 4)    // 4,800,000 floats
#define NUM_OFF    VOX_ELEMS                        // 12,000 int32
#define COOR_OFF   (NUM_OFF + MAX_VOXELS)           // 36,000 int32
#define VOXNUM_OFF (COOR_OFF + MAX_VOXELS * 3)      // 1 int32

// Scratch: module-scope device globals (re-initialized every launch -> graph-safe)
__device__ int                g_first[NUM_CELLS];
__device__ int                g_cellvox[NUM_CELLS];
__device__ unsigned long long g_mark[WORDS_MAX];
__device__ int                g_wordpref[WORDS_MAX];
__device__ unsigned int       g_counts[(size_t)MAX_CHUNKS * NUM_CELLS];  // 32 MB
__device__ int                g_cutoff;
__device__ int                g_total;

__device__ __forceinline__ bool cell_of(const float4& p, int& key) {
  // Must mirror jnp.floor((x - lo) / vs) in float32 exactly: subtract, divide, floor.
  float fx = floorf((p.x - 0.0f)      / 0.16f);
  float fy = floorf((p.y - (-10.24f)) / 0.16f);
  float fz = floorf((p.z - (-3.0f))   / 4.0f);
  if (!(fx >= 0.0f && fx < (float)GX)) return false;
  if (!(fy >= 0.0f && fy < (float)GY)) return false;
  if (!(fz >= 0.0f && fz < (float)GZ)) return false;
  int cx = (int)fx, cy = (int)fy, cz = (int)fz;
  key = (cz * GY + cy) * GX + cx;
  return true;
}

// K0: zero/initialize everything (output buffer + all scratch) each call.
// Wide b128 stores for the two big clears (output ~19.4 MB, counts ~32 MB).
__global__ void k0_init(int* outI, long long out_n, int N, int nChunks) {
  long long i = (long long)blockIdx.x * blockDim.x + threadIdx.x;
  long long stride = (long long)gridDim.x * blockDim.x;
  int4 z4 = make_int4(0, 0, 0, 0);

  long long out4 = out_n >> 2;
  int4* o4 = (int4*)outI;
  for (long long j = i; j < out4; j += stride) o4[j] = z4;
  for (long long j = (out4 << 2) + i; j < out_n; j += stride) outI[j] = 0;

  for (long long j = i; j < NUM_CELLS; j += stride) { g_first[j] = INT_MAX; g_cellvox[j] = -1; }

  long long words = (N + 63) >> 6;
  for (long long j = i; j < words; j += stride) g_mark[j] = 0ull;

  long long cn4 = ((long long)nChunks * NUM_CELLS) >> 2;  // NUM_CELLS multiple of 4
  int4* c4 = (int4*)g_counts;
  for (long long j = i; j < cn4; j += stride) c4[j] = z4;

  if (i == 0) { g_cutoff = N; g_total = 0; }
}

// K1: first[cell] = min point index (order-independent atomicMin).
__global__ void k1_first(const float4* __restrict__ pts, int N) {
  int i = blockIdx.x * blockDim.x + threadIdx.x;
  if (i >= N) return;
  __builtin_prefetch(((const char*)(pts + i)) + 8192, 0, 1);  // global_prefetch_b8
  float4 p = pts[i];
  int key;
  if (cell_of(p, key)) atomicMin(&g_first[key], i);
}

// K2: mark each occupied cell's first index in a bit array over point indices.
__global__ void k2_mark(int N) {
  int c = blockIdx.x * blockDim.x + threadIdx.x;
  if (c >= NUM_CELLS) return;
  int f = g_first[c];
  if (f < N) atomicOr(&g_mark[f >> 6], 1ull << (f & 63));
}

// K3: single-block exclusive scan of word popcounts -> wordPrefix; total -> voxel_num.
__global__ void k3_scan(int* outI, int N) {
  __shared__ int s[1024];
  int tid = threadIdx.x;
  int words = (N + 63) >> 6;
  int carry = 0;
  for (int base = 0; base < words; base += 1024) {
    int w = base + tid;
    int v = (w < words) ? __popcll(g_mark[w]) : 0;
    s[tid] = v;
    __syncthreads();
    for (int off = 1; off < 1024; off <<= 1) {
      int add = (tid >= off) ? s[tid - off] : 0;
      __syncthreads();
      s[tid] += add;
      __syncthreads();
    }
    if (w < words) g_wordpref[w] = carry + s[tid] - v;   // exclusive
    __syncthreads();
    carry += s[1023];
    __syncthreads();
  }
  if (tid == 0) {
    g_total = carry;
    outI[VOXNUM_OFF] = (carry < MAX_VOXELS) ? carry : MAX_VOXELS;
  }
}

// K4: rank each occupied cell's first among all firsts; assign voxel id, coors, cutoff.
__global__ void k4_rank(int* outI, int N) {
  int c = blockIdx.x * blockDim.x + threadIdx.x;
  if (c >= NUM_CELLS) return;
  int f = g_first[c];
  if (f >= N) return;
  unsigned long long w = g_mark[f >> 6];
  unsigned long long lm = w & ((1ull << (f & 63)) - 1ull);
  int r = g_wordpref[f >> 6] + __popcll(lm);
  if (r < MAX_VOXELS) {
    g_cellvox[c] = r;
    int cz = c / (GX * GY);
    int cy = (c / GX) % GY;
    int cx = c % GX;
    outI[COOR_OFF + 3 * r + 0] = cz;
    outI[COOR_OFF + 3 * r + 1] = cy;
    outI[COOR_OFF + 3 * r + 2] = cx;
  } else if (r == MAX_VOXELS) {
    g_cutoff = f;  // unique writer: ranks are strictly monotone in f
  }
}

// K5: per-chunk per-cell histogram of valid points (idx < cutoff).
__global__ void k5_hist(const float4* __restrict__ pts, int N) {
  int c = blockIdx.x;
  int cut = g_cutoff;
  int base = c * CHUNK_SIZE;
  if (base >= cut) return;
  unsigned int* row = g_counts + (size_t)c * NUM_CELLS;
  for (int j = threadIdx.x; j < CHUNK_SIZE; j += blockDim.x) {
    int i = base + j;
    if (i >= N || i >= cut) continue;
    float4 p = pts[i];
    int key;
    if (cell_of(p, key)) atomicAdd(&row[key], 1u);
  }
}

// K6: per-cell exclusive scan across chunks (counts row c -> prefix of chunks < c).
__global__ void k6_colscan(int nChunks) {
  int cell = blockIdx.x * blockDim.x + threadIdx.x;
  if (cell >= NUM_CELLS) return;
  unsigned int run = 0;
  for (int c = 0; c < nChunks; ++c) {
    size_t idx = (size_t)c * NUM_CELLS + cell;
    unsigned int v = g_counts[idx];
    g_counts[idx] = run;
    run += v;
  }
}

// K7: deterministic index-ordered slot assignment + scatter into voxels.
// Point data is staged global -> LDS with the CDNA5 async-copy path
// (global_load_async_to_lds_b128 + s_wait_asynccnt), one lane-owned slot
// per thread, so no cross-lane synchronization is needed for the staging.
__global__ void k7_slots(const float4* __restrict__ pts, float* out, int N) {
  __shared__ float4 sbuf[256];
  __shared__ int skey[256];
  int tid = threadIdx.x;
  int c = blockIdx.x;
  int base = c * CHUNK_SIZE;
  int cut = g_cutoff;
  if (base >= cut) return;  // uniform exit: whole block idle
  unsigned int* row = g_counts + (size_t)c * NUM_CELLS;
  int* outI = (int*)out;
  float4* outV = (float4*)out;
  // Low 32 bits of a flat shared-aperture address are the LDS byte offset.
  unsigned ldsa = (unsigned)(size_t)(const void*)(sbuf + tid);

  for (int batch = 0; batch < CHUNK_SIZE / 256; ++batch) {
    int i = base + batch * 256 + tid;
    bool want = (i < N) && (i < cut);
    if (want) {
      unsigned long long ga = (unsigned long long)(size_t)(const void*)(pts + i);
      asm volatile("global_load_async_to_lds_b128 %0, %1, off"
                   :: "v"(ldsa), "v"(ga) : "memory");
    }
    asm volatile("s_wait_asynccnt 0x0" ::: "memory");

    bool act = false;
    int key = 0x40000000 + tid;  // unique sentinel, never equals a real cell key
    float4 p = make_float4(0.f, 0.f, 0.f, 0.f);
    if (want) {
      p = sbuf[tid];             // ds_load_b128 of this lane's staged point
      int k2;
      if (cell_of(p, k2)) { key = k2; act = true; }
    }
    skey[tid] = key;
    __syncthreads();

    int r = 0, mtot = 0;
    unsigned int basecnt = 0;
    if (act) {
      for (int j = 0; j < 256; ++j) {
        bool m = (skey[j] == key);
        mtot += m ? 1 : 0;
        r += (m && j < tid) ? 1 : 0;   // index-ordered intra-batch rank
      }
      basecnt = row[key];              // running per-cell count before this batch
    }
    __syncthreads();                   // all reads of row[] complete before the write
    if (act && r == mtot - 1) row[key] = basecnt + (unsigned)mtot;  // unique writer/cell
    if (act) {
      int slot = (int)basecnt + r;
      int vox = g_cellvox[key];
      if (vox >= 0 && slot < MAX_POINTS) {
        outV[(size_t)vox * MAX_POINTS + slot] = p;     // global_store_b128, unique target
        atomicAdd(&outI[NUM_OFF + vox], 1);
      }
    }
    __syncthreads();                   // publish row[] update before next batch
  }
}

extern "C" void kernel_launch(void* const* d_in, const int* in_sizes, int n_in,
                              void* d_out, int out_size, void* d_ws, size_t ws_size,
                              hipStream_t stream) {
  const float4* pts = (const float4*)d_in[0];
  int N = in_sizes[0] / 4;
  int nChunks = (N + CHUNK_SIZE - 1) / CHUNK_SIZE;
  if (nChunks > MAX_CHUNKS) nChunks = MAX_CHUNKS;
  int* outI = (int*)d_out;
  float* outF = (float*)d_out;

  k0_init<<<2048, 256, 0, stream>>>(outI, (long long)out_size, N, nChunks);
  k1_first<<<(N + 255) / 256, 256, 0, stream>>>(pts, N);
  k2_mark<<<(NUM_CELLS + 255) / 256, 256, 0, stream>>>(N);
  k3_scan<<<1, 1024, 0, stream>>>(outI, N);
  k4_rank<<<(NUM_CELLS + 255) / 256, 256, 0, stream>>>(outI, N);
  k5_hist<<<nChunks, 256, 0, stream>>>(pts, N);
  k6_colscan<<<(NUM_CELLS + 255) / 256, 256, 0, stream>>>(nChunks);
  k7_slots<<<nChunks, 256, 0, stream>>>(pts, outF, N);
}